// BertSelfAttention_257698038467
// MI455X (gfx1250) — compile-verified
//
#include <hip/hip_runtime.h>
#include <math.h>

typedef float v2f __attribute__((ext_vector_type(2)));
typedef float v8f __attribute__((ext_vector_type(8)));
typedef unsigned int u32x4 __attribute__((ext_vector_type(4)));
typedef int i32x4 __attribute__((ext_vector_type(4)));
typedef int i32x8 __attribute__((ext_vector_type(8)));

#define BS 4
#define SEQ 2048
#define DIM 256
#define HEADS 4
#define HDIM 1024
#define LN_EPS 1e-5f
#define KC 32   // K-chunk staged through LDS by the TDM

// ---------------------------------------------------------------------------
// Issue one TDM 2D tile load: global (tile_d1 rows x tile_d0 contiguous
// elements, row stride stride0 elements, fp32) -> LDS at lds_off, packed
// row-major [tile_d1][tile_d0]. D# per CDNA5 ISA ch.8 (groups 0/1; rest zero).
// Tensor dims == tile dims (tiles are always fully in-bounds here).
// ---------------------------------------------------------------------------
__device__ __forceinline__ void tdm_load_2d_f32(
    unsigned lds_off, const float* gptr,
    unsigned tile_d0, unsigned tile_d1, unsigned long long stride0)
{
  const unsigned long long ga = (unsigned long long)(uintptr_t)gptr;
  u32x4 g0;
  g0[0] = 1u;                                                // count=1, user D#
  g0[1] = lds_off;                                           // lds_addr (bytes)
  g0[2] = (unsigned)(ga & 0xffffffffu);                      // global_addr lo
  g0[3] = (unsigned)((ga >> 32) & 0x01ffffffu) | (2u << 30); // addr hi | type=2
  i32x8 g1;
  g1[0] = (int)(2u << 16);                                   // data_size=4B
  g1[1] = (int)((tile_d0 & 0xffffu) << 16);                  // tensor_dim0 lo
  g1[2] = (int)((tile_d0 >> 16) | ((tile_d1 & 0xffffu) << 16)); // td0 hi|td1 lo
  g1[3] = (int)((tile_d1 >> 16) | (tile_d0 << 16));          // td1 hi|tile_dim0
  g1[4] = (int)(tile_d1 & 0xffffu);                          // tile_dim1, td2=0
  g1[5] = (int)(stride0 & 0xffffffffu);                      // dim0 stride lo
  g1[6] = (int)((stride0 >> 32) & 0xffffu);                  // dim0 stride hi
  g1[7] = 0;
  const i32x4 gz4 = {0, 0, 0, 0};                            // groups 2/3 unused
  const i32x8 gz8 = {0, 0, 0, 0, 0, 0, 0, 0};
  __builtin_amdgcn_tensor_load_to_lds(g0, g1, gz4, gz4, gz8, 0);
}

// ---------------------------------------------------------------------------
// Row-major FP32 WMMA GEMM:  C[z] = alpha * A[z] @ op(B[z])
// V_WMMA_F32_16X16X4_F32; block = 256 threads = 8 waves; block tile 256x64,
// each wave 32x64 (2 A frags x 4 B frags = 8 accumulators, so every staged
// B fragment feeds two WMMAs). B panels (64 x KC) are DMA'd into LDS by the
// Tensor Data Mover, double-buffered: the TDM fetches chunk c+1 while all
// waves compute chunk c. A fragments for the whole chunk are preloaded into
// registers before the publish barrier so their latency overlaps the barrier
// and the inner loop is pure LDS+WMMA.
// Fragment layouts per ISA 7.12.2 (A 16x4 / B 4x16 / C 16x16).
// Grids must divide M (256) and N (64) exactly (barriers are block-uniform).
// ---------------------------------------------------------------------------
template <bool TRANS_B>
__global__ __launch_bounds__(256) void wmma_gemm_f32(
    const float* __restrict__ Abase, long long sAz, int lda,
    const float* __restrict__ Bbase, long long sBz, int ldb,
    float* __restrict__ Cbase, long long sCz, int ldc,
    int M, int N, int K, float alpha)
{
  __shared__ float bsm[2][64 * KC];   // 2 x 8 KB double-buffered B panel

  const int z = blockIdx.z;
  const float* A = Abase + (long long)z * sAz;
  const float* B = Bbase + (long long)z * sBz;
  float* C = Cbase + (long long)z * sCz;

  const int wave  = threadIdx.x >> 5;   // 0..7
  const int lane  = threadIdx.x & 31;
  const int lr    = lane & 15;          // row (A) / col (B,C) within 16
  const int khalf = (lane >> 4) * 2;    // K sub-pair select: 0 or 2

  const int m0 = blockIdx.y * 256 + wave * 32;
  const int n0 = blockIdx.x * 64;

  const float* arow0 = A + (long long)(m0 + lr) * lda + khalf;
  const float* arow1 = arow0 + (long long)16 * lda;

  v8f acc0[4] = {v8f{}, v8f{}, v8f{}, v8f{}};
  v8f acc1[4] = {v8f{}, v8f{}, v8f{}, v8f{}};

  const int nch = K / KC;

  // Prologue: DMA chunk 0 into buffer 0.
  if (wave == 0) {
    if (TRANS_B) {
      tdm_load_2d_f32((unsigned)(uintptr_t)&bsm[0][0],
                      B + (long long)n0 * ldb, KC, 64,
                      (unsigned long long)ldb);
    } else {
      tdm_load_2d_f32((unsigned)(uintptr_t)&bsm[0][0],
                      B + n0, 64, KC, (unsigned long long)ldb);
    }
  }

  for (int c = 0; c < nch; ++c) {
    const int kbase = c * KC;

    // Preload this chunk's A fragments (one clause of global_load_b64);
    // latency overlaps the tensor-wait and the publish barrier below.
    v2f a0r[KC / 4], a1r[KC / 4];
#pragma unroll
    for (int kk = 0; kk < KC; kk += 4) {
      a0r[kk >> 2].x = arow0[kbase + kk];
      a0r[kk >> 2].y = arow0[kbase + kk + 1];
      a1r[kk >> 2].x = arow1[kbase + kk];
      a1r[kk >> 2].y = arow1[kbase + kk + 1];
    }

    if (wave == 0) __builtin_amdgcn_s_wait_tensorcnt(0);  // buf[c&1] complete
    __syncthreads();   // publish buf[c&1]; all waves done reading buf[(c+1)&1]
    if (wave == 0 && c + 1 < nch) {
      const int kn = (c + 1) * KC;
      if (TRANS_B) {
        tdm_load_2d_f32((unsigned)(uintptr_t)&bsm[(c + 1) & 1][0],
                        B + (long long)n0 * ldb + kn, KC, 64,
                        (unsigned long long)ldb);
      } else {
        tdm_load_2d_f32((unsigned)(uintptr_t)&bsm[(c + 1) & 1][0],
                        B + (long long)kn * ldb + n0, 64, KC,
                        (unsigned long long)ldb);
      }
    }

    const float* bp = &bsm[c & 1][0];
#pragma unroll
    for (int kk = 0; kk < KC; kk += 4) {
      const v2f a0 = a0r[kk >> 2];
      const v2f a1 = a1r[kk >> 2];
#pragma unroll
      for (int nt = 0; nt < 4; ++nt) {
        const int nl = nt * 16 + lr;
        v2f b;
        if (TRANS_B) {
          b.x = bp[nl * KC + kk + khalf];
          b.y = bp[nl * KC + kk + khalf + 1];
        } else {
          b.x = bp[(kk + khalf) * 64 + nl];
          b.y = bp[(kk + khalf + 1) * 64 + nl];
        }
        acc0[nt] = __builtin_amdgcn_wmma_f32_16x16x4_f32(
            false, a0, false, b, (short)0, acc0[nt], false, false);
        acc1[nt] = __builtin_amdgcn_wmma_f32_16x16x4_f32(
            false, a1, false, b, (short)0, acc1[nt], false, false);
      }
    }
  }

  const int rowbase = m0 + (lane >> 4) * 8;
#pragma unroll
  for (int nt = 0; nt < 4; ++nt) {
    const int n = n0 + nt * 16 + lr;
#pragma unroll
    for (int r = 0; r < 8; ++r) {
      C[(long long)(rowbase + r) * ldc + n] = acc0[nt][r] * alpha;
      C[(long long)(rowbase + 16 + r) * ldc + n] = acc1[nt][r] * alpha;
    }
  }
}

// ---------------------------------------------------------------------------
// In-place softmax over rows of length 2048 (one 256-thread block per row).
// 8 elements/thread cached in registers: single global read + single write.
// ---------------------------------------------------------------------------
__global__ __launch_bounds__(256) void softmax2048(float* __restrict__ P)
{
  float* p = P + (long long)blockIdx.x * 2048;
  const int tid = threadIdx.x;
  __shared__ float red[256];

  float v[8];
  float m = -3.402823466e38f;
#pragma unroll
  for (int i = 0; i < 8; ++i) {
    v[i] = p[tid + i * 256];
    m = fmaxf(m, v[i]);
  }
  red[tid] = m;
  __syncthreads();
  for (int s = 128; s > 0; s >>= 1) {
    if (tid < s) red[tid] = fmaxf(red[tid], red[tid + s]);
    __syncthreads();
  }
  m = red[0];
  __syncthreads();

  float sum = 0.f;
#pragma unroll
  for (int i = 0; i < 8; ++i) {
    v[i] = __expf(v[i] - m);
    sum += v[i];
  }
  red[tid] = sum;
  __syncthreads();
  for (int s = 128; s > 0; s >>= 1) {
    if (tid < s) red[tid] += red[tid + s];
    __syncthreads();
  }
  const float inv = 1.f / red[0];
#pragma unroll
  for (int i = 0; i < 8; ++i) p[tid + i * 256] = v[i] * inv;
}

// ---------------------------------------------------------------------------
// out[row] = LayerNorm(t[row] + x[row]) * gamma + beta   (row length = 256)
// ---------------------------------------------------------------------------
__global__ __launch_bounds__(256) void bias_residual_ln(
    const float* __restrict__ t, const float* __restrict__ x,
    const float* __restrict__ gamma, const float* __restrict__ beta,
    float* __restrict__ out)
{
  const long long row = blockIdx.x;
  const int tid = threadIdx.x;
  __shared__ float red[256];

  const float v = t[row * DIM + tid] + x[row * DIM + tid];

  red[tid] = v;
  __syncthreads();
  for (int s = 128; s > 0; s >>= 1) {
    if (tid < s) red[tid] += red[tid + s];
    __syncthreads();
  }
  const float mu = red[0] * (1.0f / DIM);
  __syncthreads();

  const float dv = v - mu;
  red[tid] = dv * dv;
  __syncthreads();
  for (int s = 128; s > 0; s >>= 1) {
    if (tid < s) red[tid] += red[tid + s];
    __syncthreads();
  }
  const float var = red[0] * (1.0f / DIM);

  out[row * DIM + tid] = dv * rsqrtf(var + LN_EPS) * gamma[tid] + beta[tid];
}

// ---------------------------------------------------------------------------
extern "C" void kernel_launch(void* const* d_in, const int* in_sizes, int n_in,
                              void* d_out, int out_size, void* d_ws, size_t ws_size,
                              hipStream_t stream)
{
  const float* x     = (const float*)d_in[0];  // (4,2048,256)
  const float* Wq    = (const float*)d_in[1];  // (256,1024)
  const float* Wk    = (const float*)d_in[2];  // (256,1024)
  const float* Wv    = (const float*)d_in[3];  // (1024,256)
  const float* gamma = (const float*)d_in[4];  // (256,)
  const float* beta  = (const float*)d_in[5];  // (256,)

  float* out   = (float*)d_out;                         // (4,2048,256)
  float* probs = out + (long long)BS * SEQ * DIM;       // (4,4,2048,2048)

  float* ws = (float*)d_ws;
  float* Qp  = ws;                    // (8192,1024) = 32 MB
  float* Kp  = ws + 8388608;          // (8192,1024) = 32 MB
  float* ctx = Qp;                    // reuse: Qp dead after scores GEMM
  float* tmp = Kp;                    // reuse: Kp dead after scores GEMM

  const dim3 blk(256);
  const long long SD = (long long)SEQ * DIM;   // 524288 per-head slab
  const long long SS = (long long)SEQ * SEQ;   // 4194304 per-head probs
  const int MR = BS * SEQ;                     // 8192 rows

  // 1) Projections: (8192x256)@(256x1024)
  wmma_gemm_f32<false><<<dim3(HDIM / 64, MR / 256, 1), blk, 0, stream>>>(
      x, 0, DIM, Wq, 0, HDIM, Qp, 0, HDIM, MR, HDIM, DIM, 1.0f);
  wmma_gemm_f32<false><<<dim3(HDIM / 64, MR / 256, 1), blk, 0, stream>>>(
      x, 0, DIM, Wk, 0, HDIM, Kp, 0, HDIM, MR, HDIM, DIM, 1.0f);

  // 2) Scores per (b,h) slab z: Qh(2048x256) @ Kh(2048x256)^T / sqrt(256)
  //    written pre-softmax into the probs region of d_out.
  wmma_gemm_f32<true><<<dim3(SEQ / 64, SEQ / 256, BS * HEADS), blk, 0, stream>>>(
      Qp, SD, DIM, Kp, SD, DIM, probs, SS, SEQ, SEQ, SEQ, DIM, 1.0f / 16.0f);

  // 3) Softmax in place over 32768 rows of 2048
  softmax2048<<<dim3(BS * HEADS * SEQ), blk, 0, stream>>>(probs);

  // 4) Context: per (b,h): probs[z](2048x2048) @ x[b](2048x256),
  //    scattered into ctx (bs, s, h*d) at column offset h*256
  for (int b = 0; b < BS; ++b) {
    for (int h = 0; h < HEADS; ++h) {
      const int zz = b * HEADS + h;
      wmma_gemm_f32<false><<<dim3(DIM / 64, SEQ / 256, 1), blk, 0, stream>>>(
          probs + (long long)zz * SS, 0, SEQ,
          x + (long long)b * SD, 0, DIM,
          ctx + (long long)b * SEQ * HDIM + (long long)h * DIM, 0, HDIM,
          SEQ, DIM, SEQ, 1.0f);
    }
  }

  // 5) Output projection: ctx(8192x1024) @ Wv(1024x256) -> tmp
  wmma_gemm_f32<false><<<dim3(DIM / 64, MR / 256, 1), blk, 0, stream>>>(
      ctx, 0, HDIM, Wv, 0, DIM, tmp, 0, DIM, MR, DIM, HDIM, 1.0f);

  // 6) Residual + LayerNorm -> out
  bias_residual_ln<<<dim3(MR), blk, 0, stream>>>(tmp, x, gamma, beta, out);
}